// BeliefPropagation_59382217834766
// MI455X (gfx1250) — compile-verified
//
#include <hip/hip_runtime.h>
#include <math.h>

#define Q 8
#define TPB 256
#define MAX_ITER 10

// ---- CDNA5 async-to-LDS probe (compiles either way) -------------------------
#if defined(__has_builtin)
#  if __has_builtin(__builtin_amdgcn_global_load_async_to_lds_b32)
#    define HAVE_ASYNC_LDS 1
#  else
#    define HAVE_ASYNC_LDS 0
#  endif
#  if __has_builtin(__builtin_amdgcn_s_wait_asynccnt)
#    define HAVE_WAIT_ASYNC 1
#  else
#    define HAVE_WAIT_ASYNC 0
#  endif
#else
#  define HAVE_ASYNC_LDS 0
#  define HAVE_WAIT_ASYNC 0
#endif

#if HAVE_ASYNC_LDS
__device__ __forceinline__ void wait_asynccnt0() {
#if HAVE_WAIT_ASYNC
  __builtin_amdgcn_s_wait_asynccnt(0);
#else
  asm volatile("s_wait_asynccnt 0" ::: "memory");
#endif
}
#endif

// ---- helpers ---------------------------------------------------------------
__device__ __forceinline__ void load8(const float* __restrict__ p, float* x) {
  const float4* q = (const float4*)p;
  float4 a = q[0], b = q[1];
  x[0]=a.x; x[1]=a.y; x[2]=a.z; x[3]=a.w; x[4]=b.x; x[5]=b.y; x[6]=b.z; x[7]=b.w;
}
__device__ __forceinline__ void store8(float* __restrict__ p, const float* x) {
  float4* q = (float4*)p;
  q[0] = make_float4(x[0], x[1], x[2], x[3]);
  q[1] = make_float4(x[4], x[5], x[6], x[7]);
}
__device__ __forceinline__ void softmax8(float* x) {
  float mx = x[0];
#pragma unroll
  for (int k = 1; k < Q; ++k) mx = fmaxf(mx, x[k]);
  float s = 0.f;
#pragma unroll
  for (int k = 0; k < Q; ++k) { x[k] = __expf(x[k] - mx); s += x[k]; }
  float inv = 1.0f / s;
#pragma unroll
  for (int k = 0; k < Q; ++k) x[k] *= inv;
}
// wave32 shuffle reduction (CDNA5 is wave32: width MUST be 32, not 64)
__device__ __forceinline__ float wavesum32(float v) {
#pragma unroll
  for (int off = 16; off > 0; off >>= 1) v += __shfl_xor(v, off, 32);
  return v;
}
// block reduce -> one hardware f32 atomic per block. ALL threads must call.
__device__ __forceinline__ void block_atomic_acc(float v, float* gaddr, float* red) {
  v = wavesum32(v);
  int lane = threadIdx.x & 31;
  int wave = threadIdx.x >> 5;
  if (lane == 0) red[wave] = v;
  __syncthreads();
  if (wave == 0) {
    float x = (lane < (int)(blockDim.x >> 5)) ? red[lane] : 0.0f;
    x = wavesum32(x);
    if (lane == 0) unsafeAtomicAdd(gaddr, x);   // global_atomic_add_f32 (no CAS loop)
  }
  __syncthreads();
}

// ---- kernels ---------------------------------------------------------------
__global__ __launch_bounds__(TPB) void k_zero(float* __restrict__ p, int n) {
  int i = blockIdx.x * TPB + threadIdx.x;
  if (i < n) p[i] = 0.0f;
}

__global__ __launch_bounds__(TPB) void k_rownorm(const float* __restrict__ in,
                                                 float* __restrict__ out, int rows) {
  int r = blockIdx.x * TPB + threadIdx.x;
  if (r >= rows) return;
  float x[Q];
  load8(in + (size_t)r * Q, x);
  float s = 0.f;
#pragma unroll
  for (int k = 0; k < Q; ++k) s += x[k];
  float inv = 1.0f / s;
#pragma unroll
  for (int k = 0; k < Q; ++k) x[k] *= inv;
  store8(out + (size_t)r * Q, x);
}

__global__ __launch_bounds__(TPB) void k_colsum(const float* __restrict__ mat,
                                                float* __restrict__ colsum, int rows) {
  __shared__ float red[TPB / 32];
  int r = blockIdx.x * TPB + threadIdx.x;
  float p[Q];
  if (r < rows) load8(mat + (size_t)r * Q, p);
  else {
#pragma unroll
    for (int k = 0; k < Q; ++k) p[k] = 0.f;
  }
#pragma unroll
  for (int k = 0; k < Q; ++k) block_atomic_acc(p[k], colsum + k, red);
}

// edge pass 1: logterm = log1p(msg*ew) written IN PLACE over msg,
// scatter-add into node_in via hardware f32 atomics.
__global__ __launch_bounds__(TPB) void k_edge1(float* __restrict__ msg,
                                               const int* __restrict__ dst,
                                               const float* __restrict__ beta,
                                               float* __restrict__ node_in, int E) {
  int e = blockIdx.x * TPB + threadIdx.x;
  bool act = e < E;
#if HAVE_ASYNC_LDS
  __shared__ int s_dst[TPB];
  if (act) {
    // builtin signature (probed): (addrspace(1) int* /*non-const*/, addrspace(3) int*, imm, cpol)
    typedef __attribute__((address_space(1))) int gint_t;
    typedef __attribute__((address_space(3))) int lint_t;
    __builtin_amdgcn_global_load_async_to_lds_b32(
        (gint_t*)(dst + e), (lint_t*)&s_dst[threadIdx.x], 0, 0);
  }
  wait_asynccnt0();
  __syncthreads();
#endif
  if (!act) return;
  float b  = beta[0];
  float ew = expf(b) - 1.0f;
  float x[Q];
  float* row = msg + (size_t)e * Q;
  __builtin_prefetch(row, 0, 0);               // global_prefetch_b8
  load8(row, x);
#pragma unroll
  for (int k = 0; k < Q; ++k) x[k] = log1pf(x[k] * ew);
  store8(row, x);
#if HAVE_ASYNC_LDS
  int d = s_dst[threadIdx.x];
#else
  int d = dst[e];
#endif
  float* nb = node_in + (size_t)d * Q;
#pragma unroll
  for (int k = 0; k < Q; ++k) unsafeAtomicAdd(nb + k, x[k]);
}

// node pass: psi = softmax(node_in + h); accumulate psi column sums for next h
__global__ __launch_bounds__(TPB) void k_node(const float* __restrict__ node_in,
                                              const float* __restrict__ hptr,
                                              float* __restrict__ psi,
                                              float* __restrict__ colsum, int N) {
  __shared__ float red[TPB / 32];
  int n = blockIdx.x * TPB + threadIdx.x;
  float p[Q];
  if (n < N) {
    load8(node_in + (size_t)n * Q, p);
#pragma unroll
    for (int k = 0; k < Q; ++k) p[k] += hptr[k];
    softmax8(p);
    store8(psi + (size_t)n * Q, p);
  } else {
#pragma unroll
    for (int k = 0; k < Q; ++k) p[k] = 0.f;
  }
#pragma unroll
  for (int k = 0; k < Q; ++k) block_atomic_acc(p[k], colsum + k, red);
}

// edge pass 2: rev is the involution e<->e+m, so one thread handles BOTH
// directions of an undirected edge -> in-place msg update with no race.
__global__ __launch_bounds__(TPB) void k_edge2(float* __restrict__ msg,
                                               const int* __restrict__ src,
                                               const int* __restrict__ dst,
                                               const float* __restrict__ node_in,
                                               const float* __restrict__ hptr, int mcnt) {
  int e = blockIdx.x * TPB + threadIdx.x;
  if (e >= mcnt) return;
  __builtin_prefetch(msg + (size_t)(e + mcnt) * Q, 0, 0);
  int u = src[e], v = dst[e];
  float lte[Q], ltr[Q], a[Q], c[Q];
  load8(msg + (size_t)e * Q, lte);             // logterm[e]   (u->v)
  load8(msg + (size_t)(e + mcnt) * Q, ltr);    // logterm[e+m] (v->u)
  load8(node_in + (size_t)u * Q, a);           // node_in[u]
  load8(node_in + (size_t)v * Q, c);           // node_in[v]
#pragma unroll
  for (int k = 0; k < Q; ++k) {
    float h = hptr[k];
    a[k] = a[k] - ltr[k] + h;                  // new_msg[e]   = softmax(ni[src[e]] - lt[rev[e]] + h)
    c[k] = c[k] - lte[k] + h;                  // new_msg[e+m]
  }
  softmax8(a);
  softmax8(c);
  store8(msg + (size_t)e * Q, a);
  store8(msg + (size_t)(e + mcnt) * Q, c);
}

__global__ void k_update_h(const float* __restrict__ beta, float* __restrict__ h,
                           float* __restrict__ colsum, float mean_w) {
  int k = threadIdx.x;
  if (k < Q) {
    float b = beta[0];
    h[k] = -b * mean_w * colsum[k];
    colsum[k] = 0.0f;                          // ready for next iteration
  }
}

__global__ __launch_bounds__(TPB) void k_entropy(const float* __restrict__ psi,
                                                 float* __restrict__ ent_acc, int N) {
  __shared__ float red[TPB / 32];
  int n = blockIdx.x * TPB + threadIdx.x;
  float s = 0.f;
  if (n < N) {
    float p[Q];
    load8(psi + (size_t)n * Q, p);
#pragma unroll
    for (int k = 0; k < Q; ++k) s += p[k] * logf(p[k] + 1e-12f);
  }
  block_atomic_acc(s, ent_acc, red);
}

__global__ __launch_bounds__(TPB) void k_edge_metrics(const float* __restrict__ psi,
                                                      const int* __restrict__ src,
                                                      const int* __restrict__ dst,
                                                      float* __restrict__ reg_acc,
                                                      float* __restrict__ delta_acc, int mcnt) {
  __shared__ float red[TPB / 32];
  int e = blockIdx.x * TPB + threadIdx.x;
  float r = 0.f, dlt = 0.f;
  if (e < mcnt) {
    int u = src[e], v = dst[e];
    float pu[Q], pv[Q];
    load8(psi + (size_t)u * Q, pu);
    load8(psi + (size_t)v * Q, pv);
    int au = 0, av = 0;
    float mu = pu[0], mv = pv[0];
#pragma unroll
    for (int k = 1; k < Q; ++k) {              // first-max tie rule == jnp.argmax
      if (pu[k] > mu) { mu = pu[k]; au = k; }
      if (pv[k] > mv) { mv = pv[k]; av = k; }
    }
#pragma unroll
    for (int k = 0; k < Q; ++k) { float d = pu[k] - pv[k]; r += d * d; }
    dlt = (au == av) ? 1.0f : 0.0f;
  }
  block_atomic_acc(r, reg_acc, red);
  block_atomic_acc(dlt, delta_acc, red);
}

__global__ void k_finalize(float* __restrict__ out_scal, const float* __restrict__ accs,
                           float mean_w, int mcnt, int N) {
  if (threadIdx.x == 0 && blockIdx.x == 0) {
    float reg = accs[0] / (float)mcnt;
    float H = -accs[1] / (float)N;
    float scale = 4.0f * logf(0.5f) / logf(1.0f / (float)Q);  // = 4/3 for q=8
    out_scal[0] = reg;                                        // reg
    out_scal[1] = H * scale;                                  // entropy_loss
    out_scal[2] = accs[2] * (1.0f - mean_w) / (float)mcnt;    // modularity
  }
}

// ---- launch -----------------------------------------------------------------
extern "C" void kernel_launch(void* const* d_in, const int* in_sizes, int n_in,
                              void* d_out, int out_size, void* d_ws, size_t ws_size,
                              hipStream_t stream) {
  const float* msg_init = (const float*)d_in[0];
  const float* psi_init = (const float*)d_in[1];
  const float* beta     = (const float*)d_in[2];
  const int*   src      = (const int*)d_in[3];
  const int*   dst      = (const int*)d_in[4];

  const int E    = in_sizes[3];
  const int mcnt = E / 2;
  const int N    = in_sizes[1] / Q;
  const float mean_w = (float)((double)E / ((double)N * (double)N));

  float* out = (float*)d_out;
  float* psi = out;                               // (N,Q) lives in d_out directly

  // workspace: msg/logterm (E*Q) | node_in (N*Q) | h(8) | colsum(8) | accs(3)
  float* msg     = (float*)d_ws;
  float* node_in = msg + (size_t)E * Q;
  float* h       = node_in + (size_t)N * Q;
  float* colsum  = h + Q;
  float* accs    = colsum + Q;

  auto G = [](int n) { return (n + TPB - 1) / TPB; };

  // reset small state every launch (ws is poisoned / left dirty across calls)
  k_zero<<<1, 64, 0, stream>>>(colsum, Q + 3);    // colsum + accs are contiguous

  // init: row-normalize msg and psi; h0 = -b*mean_w*psi.sum(0)
  k_rownorm<<<G(E), TPB, 0, stream>>>(msg_init, msg, E);
  k_rownorm<<<G(N), TPB, 0, stream>>>(psi_init, psi, N);
  k_colsum<<<G(N), TPB, 0, stream>>>(psi, colsum, N);
  k_update_h<<<1, 32, 0, stream>>>(beta, h, colsum, mean_w);

  for (int it = 0; it < MAX_ITER; ++it) {
    k_zero<<<G(N * Q), TPB, 0, stream>>>(node_in, N * Q);
    k_edge1<<<G(E), TPB, 0, stream>>>(msg, dst, beta, node_in, E);
    k_node<<<G(N), TPB, 0, stream>>>(node_in, h, psi, colsum, N);     // uses OLD h
    k_edge2<<<G(mcnt), TPB, 0, stream>>>(msg, src, dst, node_in, h, mcnt); // OLD h
    k_update_h<<<1, 32, 0, stream>>>(beta, h, colsum, mean_w);        // h for next iter
  }

  // outputs: psi already in d_out; three scalars appended
  k_entropy<<<G(N), TPB, 0, stream>>>(psi, accs + 1, N);
  k_edge_metrics<<<G(mcnt), TPB, 0, stream>>>(psi, src, dst, accs + 0, accs + 2, mcnt);
  k_finalize<<<1, 1, 0, stream>>>(out + (size_t)N * Q, accs, mean_w, mcnt, N);
}